// GATSelfAttention_9629316678015
// MI455X (gfx1250) — compile-verified
//
#include <hip/hip_runtime.h>
#include <hip/hip_bf16.h>

// GAT self-attention for MI455X (gfx1250, wave32).
// Both GEMM stages use V_WMMA_F32_16X16X4_F32 (full-f32 matrix pipe).
// Kernel D caches each wave's 16x512 score panel in LDS (single adj pass).
//
// Shapes: N=32, E=512, IN=128, OUT=64, Q=128, NT=3.

typedef float v2f __attribute__((ext_vector_type(2)));
typedef float v8f __attribute__((ext_vector_type(8)));

#define GN  32
#define GE  512
#define GIN 128
#define GOUT 64
#define GQ  128
#define GNT 3

__device__ __forceinline__ v8f wmma4(v2f a, v2f b, v8f c) {
  // D = A(16x4) * B(4x16) + C, all f32.
  return __builtin_amdgcn_wmma_f32_16x16x4_f32(
      /*neg_a=*/false, a, /*neg_b=*/false, b,
      /*c_mod=*/(short)0, c, /*reuse_a=*/false, /*reuse_b=*/false);
}

// ---------------------------------------------------------------------------
// Kernel A: qg = sigmoid(relu(q @ W1) @ W2); wcomb[t,n,c] = qg[c]*a_type[t,c]
// One 128-thread block per (t, n).
// ---------------------------------------------------------------------------
__global__ void qgate_kernel(const float* __restrict__ qv,
                             const float* __restrict__ W1,
                             const float* __restrict__ W2,
                             const float* __restrict__ atype,
                             float* __restrict__ wcomb) {
  int t = blockIdx.x / GN;
  int n = blockIdx.x % GN;
  int c = threadIdx.x;           // 0..127
  __shared__ float qs[GQ];
  __shared__ float hs[2 * GOUT];
  qs[c] = qv[n * GQ + c];
  __syncthreads();
  float acc = 0.f;
  const float* w1 = W1 + (size_t)t * GQ * 2 * GOUT + c;
  #pragma unroll 4
  for (int q = 0; q < GQ; ++q) acc += qs[q] * w1[(size_t)q * 2 * GOUT];
  hs[c] = fmaxf(acc, 0.f);
  __syncthreads();
  float acc2 = 0.f;
  const float* w2 = W2 + (size_t)t * 2 * GOUT * 2 * GOUT + c;
  #pragma unroll 4
  for (int q = 0; q < 2 * GOUT; ++q) acc2 += hs[q] * w2[(size_t)q * 2 * GOUT];
  float g = 1.f / (1.f + __expf(-acc2));
  wcomb[((size_t)t * GN + n) * (2 * GOUT) + c] = g * atype[t * (2 * GOUT) + c];
}

// ---------------------------------------------------------------------------
// Kernel B: h_all[t,n,e,o] = input_state[n,e,:] @ W_type[t]  via WMMA f32.
// One wave per (t, n, e-tile of 16); 4 C tiles cover o=0..63; K loop 128/4.
// ---------------------------------------------------------------------------
__global__ void hall_kernel(const float* __restrict__ x,
                            const float* __restrict__ W,
                            float* __restrict__ hall) {
  int wave = blockIdx.x * (blockDim.x >> 5) + (threadIdx.x >> 5);
  int lane = threadIdx.x & 31;
  int t   = wave >> 10;          // 1024 waves per type
  int rem = wave & 1023;
  int n   = rem >> 5;
  int et  = rem & 31;
  int ml   = lane & 15;          // row within tile (A) / column (B,D)
  int half = lane >> 4;          // K sub-group

  // A[m][k]: lane=(k/2)*16+m, vgpr=k&1  -> this lane holds k = 2*half{,+1}
  const float* xa = x + ((size_t)(n * GE + et * 16 + ml)) * GIN + 2 * half;
  // B[k][o]: lane=(k/2)*16+o, vgpr=k&1
  const float* Wt = W + (size_t)t * GIN * GOUT + (size_t)(2 * half) * GOUT + ml;

  v8f acc0 = {}, acc1 = {}, acc2 = {}, acc3 = {};
  for (int kb = 0; kb < GIN; kb += 4) {
    v2f a = *(const v2f*)(xa + kb);          // k = kb+2*half, kb+2*half+1
    const float* wb = Wt + (size_t)kb * GOUT;
    v2f b0 = { wb[0],  wb[GOUT + 0]  };
    v2f b1 = { wb[16], wb[GOUT + 16] };
    v2f b2 = { wb[32], wb[GOUT + 32] };
    v2f b3 = { wb[48], wb[GOUT + 48] };
    acc0 = wmma4(a, b0, acc0);
    acc1 = wmma4(a, b1, acc1);
    acc2 = wmma4(a, b2, acc2);
    acc3 = wmma4(a, b3, acc3);
  }
  // D[m][o]: vgpr r -> m = r + 8*half, lane%16 -> o%16
  float* out = hall + (((size_t)(t * GN + n)) * GE + et * 16 + half * 8) * GOUT + ml;
  #pragma unroll
  for (int r = 0; r < 8; ++r) {
    out[r * GOUT + 0]  = acc0[r];
    out[r * GOUT + 16] = acc1[r];
    out[r * GOUT + 32] = acc2[r];
    out[r * GOUT + 48] = acc3[r];
  }
}

// ---------------------------------------------------------------------------
// Kernel C: s_src/s_dst[t,n,e] = <h_all[t,n,e,:], wcomb[t,n,(0|64)+ :]>.
// One wave32 per row; cross-lane shuffle reduction.
// ---------------------------------------------------------------------------
__global__ void score_kernel(const float* __restrict__ hall,
                             const float* __restrict__ wcomb,
                             float* __restrict__ ssrc,
                             float* __restrict__ sdst) {
  int row  = blockIdx.x * (blockDim.x >> 5) + (threadIdx.x >> 5);  // (t*32+n)*512+e
  int lane = threadIdx.x & 31;
  int tn = row >> 9;                   // t*32+n
  const float* h  = hall + (size_t)row * GOUT;
  const float* wc = wcomb + (size_t)tn * (2 * GOUT);
  float h0 = h[lane], h1 = h[lane + 32];
  float ps = h0 * wc[lane]      + h1 * wc[lane + 32];
  float pd = h0 * wc[64 + lane] + h1 * wc[96 + lane];
  #pragma unroll
  for (int off = 16; off; off >>= 1) {
    ps += __shfl_xor(ps, off, 32);
    pd += __shfl_xor(pd, off, 32);
  }
  if (lane == 0) { ssrc[row] = ps; sdst[row] = pd; }
}

// ---------------------------------------------------------------------------
// Kernel D: fused masked-score softmax (over i) + out = coefs^T @ (h*mask).
// One wave per (n, j-tile of 16); 2 waves per block.
// Pass 1: compute scores once (single adj read), cache 16x512 panel in LDS,
//         track per-j max.  Pass 1.5: normalizer from LDS (1 exp/elem).
// Pass 2: p = exp(sc-M)*rinv*mask[i] from LDS into WMMA A-fragments (P^T),
//         multiply against h tiles (4i x 16d) with V_WMMA_F32_16X16X4_F32.
// node_mask is folded into p (equivalent to masking h rows).
// ---------------------------------------------------------------------------
__device__ __forceinline__ float masked_score(int av, const float* __restrict__ ssrc,
                                              int n, int i,
                                              float sd0, float sd1, float sd2) {
  float sv = -3.0e38f;
  if (av > 0) {
    float s1 = ssrc[((size_t)(av - 1) * GN + n) * GE + i];
    float sdv = (av == 1) ? sd0 : ((av == 2) ? sd1 : sd2);
    float sc = s1 + sdv;
    sv = (sc > 0.f) ? sc : 0.2f * sc;   // leaky_relu(0.2)
  }
  return sv;
}

__global__ void attn_out_kernel(const int* __restrict__ adj,
                                const float* __restrict__ hall,
                                const float* __restrict__ ssrc,
                                const float* __restrict__ sdstg,
                                const float* __restrict__ mask,
                                float* __restrict__ out) {
  __shared__ float scb[2 * GE * 16];           // 64 KB: 16x512 panel per wave
  int wid  = threadIdx.x >> 5;                 // wave in block (0..1)
  int wave = blockIdx.x * 2 + wid;
  int lane = threadIdx.x & 31;
  int n  = wave >> 5;
  int jt = wave & 31;
  int jl   = lane & 15;
  int half = lane >> 4;
  int j = jt * 16 + jl;
  float* sc = scb + wid * (GE * 16);           // sc[i*16 + jl]

  float sd0 = sdstg[(0 * GN + n) * GE + j];
  float sd1 = sdstg[(1 * GN + n) * GE + j];
  float sd2 = sdstg[(2 * GN + n) * GE + j];

  const int* adjc = adj + (size_t)n * GE * GE + j;   // column j, stride GE

  // ---- pass 1: score -> LDS panel, per-j running max (each lane: i=half::2)
  float m = -3.0e38f;
  for (int i = half; i < GE; i += 2) {
    float sv = masked_score(adjc[(size_t)i * GE], ssrc, n, i, sd0, sd1, sd2);
    sc[i * 16 + jl] = sv;
    m = fmaxf(m, sv);
  }
  float M = fmaxf(m, __shfl_xor(m, 16, 32));

  // ---- pass 1.5: softmax normalizer from LDS (1 exp per element) ----
  float l = 0.f;
  for (int i = half; i < GE; i += 2) l += __expf(sc[i * 16 + jl] - M);
  l += __shfl_xor(l, 16, 32);
  float rinv = 1.0f / l;

  // ---- pass 2: P^T @ h via WMMA f32 16x16x4 ----
  const float* h2   = hall + ((size_t)(2 * GN + n)) * GE * GOUT;  // h_all[-1][n]
  const float* mrow = mask + (size_t)n * GE;
  v8f acc0 = {}, acc1 = {}, acc2 = {}, acc3 = {};
  for (int ib = 0; ib < GE; ib += 4) {
    int i0 = ib + 2 * half;
    // A = P^T: A[m=jl][k=i']: lane=(k/2)*16+m, vgpr=k&1
    v2f a;
    a.x = __expf(sc[(i0 + 0) * 16 + jl] - M) * rinv * mrow[i0 + 0];
    a.y = __expf(sc[(i0 + 1) * 16 + jl] - M) * rinv * mrow[i0 + 1];
    // B = h: B[k=i'][d]: lane=(k/2)*16+d%16, vgpr=k&1
    const float* hr = h2 + (size_t)i0 * GOUT + jl;
    v2f b0 = { hr[0],  hr[GOUT + 0]  };
    v2f b1 = { hr[16], hr[GOUT + 16] };
    v2f b2 = { hr[32], hr[GOUT + 32] };
    v2f b3 = { hr[48], hr[GOUT + 48] };
    acc0 = wmma4(a, b0, acc0);
    acc1 = wmma4(a, b1, acc1);
    acc2 = wmma4(a, b2, acc2);
    acc3 = wmma4(a, b3, acc3);
  }
  // D[m][d]: vgpr r -> j-local = r + 8*half, lane%16 -> d%16
  float* o = out + (((size_t)n * GE) + jt * 16 + half * 8) * GOUT + jl;
  #pragma unroll
  for (int r = 0; r < 8; ++r) {
    o[r * GOUT + 0]  = acc0[r];
    o[r * GOUT + 16] = acc1[r];
    o[r * GOUT + 32] = acc2[r];
    o[r * GOUT + 48] = acc3[r];
  }
}

// ---------------------------------------------------------------------------
extern "C" void kernel_launch(void* const* d_in, const int* in_sizes, int n_in,
                              void* d_out, int out_size, void* d_ws, size_t ws_size,
                              hipStream_t stream) {
  const float* x   = (const float*)d_in[0];  // input_state (32,512,128)
  const int*   adj = (const int*)  d_in[1];  // adj (32,512,512)
  const float* qv  = (const float*)d_in[2];  // query_vec (32,128)
  const float* msk = (const float*)d_in[3];  // node_mask (32,512,1)
  const float* Wt  = (const float*)d_in[4];  // W_type (3,128,64)
  const float* at  = (const float*)d_in[5];  // a_type (3,128,1)
  const float* W1  = (const float*)d_in[6];  // qattn_W1 (3,128,128)
  const float* W2  = (const float*)d_in[7];  // qattn_W2 (3,128,128)
  float* out = (float*)d_out;                // (32,512,64)

  // Workspace layout (floats): ~12.5 MB total.
  float* hall  = (float*)d_ws;                         // 3*32*512*64 = 3,145,728
  float* wcomb = hall  + (size_t)GNT * GN * GE * GOUT; //   12,288
  float* ssrc  = wcomb + (size_t)GNT * GN * 2 * GOUT;  //   49,152
  float* sdst  = ssrc  + (size_t)GNT * GN * GE;        //   49,152

  // A: 96 blocks x 128 threads
  qgate_kernel<<<GNT * GN, 128, 0, stream>>>(qv, W1, W2, at, wcomb);
  // B: 3072 waves (t,n,etile), 4 waves/block
  hall_kernel<<<(GNT * GN * (GE / 16)) / 4, 128, 0, stream>>>(x, Wt, hall);
  // C: 49152 rows, 8 waves/block
  score_kernel<<<(GNT * GN * GE) / 8, 256, 0, stream>>>(hall, wcomb, ssrc, sdst);
  // D: 1024 waves (n,jtile), 2 waves/block (64 KB LDS score panels)
  attn_out_kernel<<<(GN * (GE / 16)) / 2, 64, 0, stream>>>(adj, hall, ssrc, sdst, msk, out);
}